// WEA_DIFF_Layer_51032801411698
// MI455X (gfx1250) — compile-verified
//
#include <hip/hip_runtime.h>
#include <math.h>

typedef __attribute__((ext_vector_type(2))) float v2f;
typedef __attribute__((ext_vector_type(8))) float v8f;

#define WMMA_F32(a,b,c) __builtin_amdgcn_wmma_f32_16x16x4_f32(false,(a),false,(b),(short)0,(c),false,false)

constexpr int B_   = 4096;
constexpr int S_   = 50;
constexpr int HID_ = 256;
constexpr int NH_  = 4;
constexpr int HD_  = 64;
constexpr int FB_  = 26;          // S/2+1
constexpr int GE_  = NH_ * FB_;   // 104 scale/bias pairs per row
constexpr float ALPHA_ = 0.3f;
constexpr int KP_  = 52;          // padded K (time / concat[R;I] freq)
constexpr int CHP_ = 264;         // padded channel stride (bank-conflict-free)

// ---------------------------------------------------------------------------
// Kernel 1: per 16 batch rows — attr mean + MLP (gelu) -> g/e modulation arrays
// ---------------------------------------------------------------------------
__global__ __launch_bounds__(256) void mlp_adapt_kernel(
    const float* __restrict__ attr,
    const float* __restrict__ w1, const float* __restrict__ b1,
    const float* __restrict__ w2, const float* __restrict__ b2,
    const float* __restrict__ bfilt, const float* __restrict__ bbias,
    float* __restrict__ gws, float* __restrict__ ews)
{
  constexpr int CST = 260;                 // padded K stride for A-fragments
  __shared__ float sCtx[16 * CST];
  __shared__ float sH[16 * CST];
  __shared__ float sAdapt[16 * 208];

  const int tid  = threadIdx.x;
  const int lane = tid & 31, wave = tid >> 5;
  const int al   = lane & 15;              // A-row / B-col within tile
  const int kh   = (lane >> 4) * 2;        // half-wave K offset
  const int mh   = (lane >> 4) * 8;        // half-wave M offset (C/D layout)
  const int b0   = blockIdx.x * 16;

  // attr context: mean over S (coalesced across channels)
  for (int i = tid; i < 16 * 256; i += 256) {
    int m = i >> 8, c = i & 255;
    const float* p = attr + ((size_t)(b0 + m) * S_) * HID_ + c;
    float s = 0.f;
    #pragma unroll 10
    for (int t = 0; t < S_; ++t) s += p[t * HID_];
    sCtx[m * CST + c] = s * (1.0f / S_);
  }
  __syncthreads();

  // GEMM1: h = gelu(ctx @ w1 + b1) : 16x256 @ 256x256, 16 N-tiles
  for (int t = wave; t < 16; t += 8) {
    const int n0 = t * 16;
    v8f c = {};
    for (int kk = 0; kk < 64; ++kk) {
      const int k = kk * 4 + kh;
      v2f a, bf;
      a.x  = sCtx[al * CST + k];
      a.y  = sCtx[al * CST + k + 1];
      bf.x = w1[(size_t)k * HID_ + n0 + al];
      bf.y = w1[(size_t)(k + 1) * HID_ + n0 + al];
      c = WMMA_F32(a, bf, c);
    }
    const int n = n0 + al;
    const float bias = b1[n];
    #pragma unroll
    for (int j = 0; j < 8; ++j) {
      float h = c[j] + bias;
      h = 0.5f * h * (1.0f + erff(h * 0.70710678118f));   // exact gelu
      sH[(j + mh) * CST + n] = h;
    }
  }
  __syncthreads();

  // GEMM2: adapt = h @ w2 + b2 : 16x256 @ 256x208, 13 N-tiles
  for (int t = wave; t < 13; t += 8) {
    const int n0 = t * 16;
    v8f c = {};
    for (int kk = 0; kk < 64; ++kk) {
      const int k = kk * 4 + kh;
      v2f a, bf;
      a.x  = sH[al * CST + k];
      a.y  = sH[al * CST + k + 1];
      bf.x = w2[(size_t)k * 208 + n0 + al];
      bf.y = w2[(size_t)(k + 1) * 208 + n0 + al];
      c = WMMA_F32(a, bf, c);
    }
    const int n = n0 + al;
    const float bias = b2[n];
    #pragma unroll
    for (int j = 0; j < 8; ++j) sAdapt[(j + mh) * 208 + n] = c[j] + bias;
  }
  __syncthreads();

  // g = base_filter * (1 + scale); e = base_bias + bias
  for (int i = tid; i < 16 * GE_; i += 256) {
    int m = i / GE_, q = i - m * GE_;   // q = nh*FB + f
    float sc = sAdapt[m * 208 + 2 * q];
    float bi = sAdapt[m * 208 + 2 * q + 1];
    gws[(size_t)(b0 + m) * GE_ + q] = bfilt[q] * (1.0f + sc);
    ews[(size_t)(b0 + m) * GE_ + q] = bbias[q] + bi;
  }
}

// ---------------------------------------------------------------------------
// Kernel 2: per batch row — DFT-as-GEMM (fwd), modulate, wavelet (in place),
// DFT-as-GEMM (inv, fused combine), LayerNorm. All LDS-resident (~146 KB).
// ---------------------------------------------------------------------------
__global__ __launch_bounds__(256) void spectral_kernel(
    const float* __restrict__ item, const float* __restrict__ cw,
    const float* __restrict__ gws,  const float* __restrict__ ews,
    const float* __restrict__ gamma, const float* __restrict__ beta,
    float* __restrict__ out)
{
  __shared__ __align__(16) float sX[KP_ * CHP_];   // item -> res (in place)
  __shared__ __align__(16) float sRI[64 * CHP_];   // [R;I] spectrum
  __shared__ float sCf[64 * KP_];                  // forward basis (freq x time)
  __shared__ float sWi[64 * KP_];                  // inverse basis (time x freq)
  __shared__ float sG[GE_], sE[GE_];

  const int tid  = threadIdx.x;
  const int lane = tid & 31, wave = tid >> 5;
  const int al   = lane & 15;
  const int kh   = (lane >> 4) * 2;
  const int mh   = (lane >> 4) * 8;
  const int b    = blockIdx.x;

  // load item[b] into LDS (float4), zero K-pad rows 50..51
  {
    const float4* src = (const float4*)(item + (size_t)b * S_ * HID_);
    for (int i = tid; i < S_ * HID_ / 4; i += 256) {
      int s = i >> 6, q = i & 63;
      *(float4*)&sX[s * CHP_ + q * 4] = src[i];
    }
    sX[50 * CHP_ + tid] = 0.f;
    sX[51 * CHP_ + tid] = 0.f;
  }
  if (tid < GE_) { sG[tid] = gws[(size_t)b * GE_ + tid];
                   sE[tid] = ews[(size_t)b * GE_ + tid]; }

  // DFT bases (ortho). fwd rows: [cos(f t); -sin(f t)]/sqrt(50)
  // inv: x[s] = (1/sqrt(50)) sum_f w_f (R' cos - I' sin), w=1 at f=0,25 else 2
  const float wstep = 6.283185307179586f / 50.f;
  const float inv   = 0.1414213562373095f;   // 1/sqrt(50)
  for (int i = tid; i < 64 * KP_; i += 256) {
    int r = i / KP_, t = i - r * KP_;
    float cf = 0.f, wi = 0.f;
    if (t < 50 && r < 52) {
      int f = (r < 26) ? r : (r - 26);
      float ang = wstep * (float)(f * t);
      cf = (r < 26) ? (cosf(ang) * inv) : (-sinf(ang) * inv);
    }
    if (r < 50) {
      int f = (t < 26) ? t : (t - 26);
      float wgt = (f == 0 || f == 25) ? 1.f : 2.f;
      float ang = wstep * (float)(f * r);
      wi = (t < 26) ? (wgt * cosf(ang) * inv) : (-wgt * sinf(ang) * inv);
    }
    sCf[i] = cf;
    sWi[i] = wi;
  }
  __syncthreads();

  // forward: RI(64x256) = Cf(64x52) @ X(52x256) ; 4 M-tiles x 16 N-tiles
  for (int t = wave; t < 64; t += 8) {
    const int m0 = (t >> 4) * 16, n0 = (t & 15) * 16;
    v8f c = {};
    for (int kk = 0; kk < 13; ++kk) {
      const int k = kk * 4 + kh;
      v2f a, bf;
      a.x  = sCf[(m0 + al) * KP_ + k];
      a.y  = sCf[(m0 + al) * KP_ + k + 1];
      bf.x = sX[k * CHP_ + n0 + al];
      bf.y = sX[(k + 1) * CHP_ + n0 + al];
      c = WMMA_F32(a, bf, c);
    }
    #pragma unroll
    for (int j = 0; j < 8; ++j) sRI[(m0 + j + mh) * CHP_ + n0 + al] = c[j];
  }
  __syncthreads();

  // modulate spectrum: R' = g*R + e ; I' = g*I   (per-head g/e)
  for (int i = tid; i < 52 * 256; i += 256) {
    int r = i >> 8, cc = i & 255;
    int head = cc >> 6;
    float v = sRI[r * CHP_ + cc];
    if (r < 26) v = sG[head * FB_ + r] * v + sE[head * FB_ + r];
    else        v = sG[head * FB_ + (r - 26)] * v;
    sRI[r * CHP_ + cc] = v;
  }
  // wavelet mix + residual, in place: sX <- 0.7*wavelet + item
  for (int i = tid; i < 25 * 256; i += 256) {
    int p = i >> 8, cc = i & 255;
    int head = cc >> 6, hd = cc & 63;
    float xe = sX[(2 * p) * CHP_ + cc];
    float xo = sX[(2 * p + 1) * CHP_ + cc];
    float ap = 0.5f * (xe + xo);
    float de = 0.5f * (xe - xo) * cw[(head * 25 + p) * HD_ + hd];
    sX[(2 * p) * CHP_ + cc]     = (1.0f - ALPHA_) * (ap + de) + xe;
    sX[(2 * p + 1) * CHP_ + cc] = (1.0f - ALPHA_) * (ap - de) + xo;
  }
  __syncthreads();

  // inverse: Xf = Wi(64x52) @ RI'(52x256); fuse res += 0.3*xf into sX
  for (int t = wave; t < 64; t += 8) {
    const int m0 = (t >> 4) * 16, n0 = (t & 15) * 16;
    v8f c = {};
    for (int kk = 0; kk < 13; ++kk) {
      const int k = kk * 4 + kh;
      v2f a, bf;
      a.x  = sWi[(m0 + al) * KP_ + k];
      a.y  = sWi[(m0 + al) * KP_ + k + 1];
      bf.x = sRI[k * CHP_ + n0 + al];
      bf.y = sRI[(k + 1) * CHP_ + n0 + al];
      c = WMMA_F32(a, bf, c);
    }
    #pragma unroll
    for (int j = 0; j < 8; ++j) {
      int srow = m0 + j + mh;
      if (srow < 50) sX[srow * CHP_ + n0 + al] += ALPHA_ * c[j];
    }
  }
  __syncthreads();

  // LayerNorm per (s) row over 256 channels: wave32 shuffle reduction
  for (int s = wave; s < 50; s += 8) {
    float v[8];
    float part = 0.f;
    #pragma unroll
    for (int q = 0; q < 8; ++q) { v[q] = sX[s * CHP_ + lane + 32 * q]; part += v[q]; }
    #pragma unroll
    for (int off = 16; off >= 1; off >>= 1) part += __shfl_xor(part, off, 32);
    float mu = part * (1.0f / 256.0f);
    float p2 = 0.f;
    #pragma unroll
    for (int q = 0; q < 8; ++q) { float d = v[q] - mu; p2 += d * d; }
    #pragma unroll
    for (int off = 16; off >= 1; off >>= 1) p2 += __shfl_xor(p2, off, 32);
    float sig = rsqrtf(p2 * (1.0f / 256.0f) + 1e-12f);
    float* po = out + ((size_t)b * S_ + s) * HID_;
    #pragma unroll
    for (int q = 0; q < 8; ++q) {
      int cc = lane + 32 * q;
      po[cc] = (v[q] - mu) * sig * gamma[cc] + beta[cc];
    }
  }
}

// ---------------------------------------------------------------------------
extern "C" void kernel_launch(void* const* d_in, const int* in_sizes, int n_in,
                              void* d_out, int out_size, void* d_ws, size_t ws_size,
                              hipStream_t stream) {
  const float* item  = (const float*)d_in[0];
  const float* attr  = (const float*)d_in[1];
  const float* cw    = (const float*)d_in[2];
  const float* bfilt = (const float*)d_in[3];
  const float* bbias = (const float*)d_in[4];
  const float* w1    = (const float*)d_in[5];
  const float* b1    = (const float*)d_in[6];
  const float* w2    = (const float*)d_in[7];
  const float* b2    = (const float*)d_in[8];
  const float* gam   = (const float*)d_in[9];
  const float* bet   = (const float*)d_in[10];

  float* gws = (float*)d_ws;                       // B * 104 floats
  float* ews = gws + (size_t)B_ * GE_;             // B * 104 floats

  mlp_adapt_kernel<<<B_ / 16, 256, 0, stream>>>(attr, w1, b1, w2, b2,
                                                bfilt, bbias, gws, ews);
  spectral_kernel<<<B_, 256, 0, stream>>>(item, cw, gws, ews, gam, bet,
                                          (float*)d_out);
}